// PillarFeatureNet2_MSG_11003706213045
// MI455X (gfx1250) — compile-verified
//
#include <hip/hip_runtime.h>

typedef float v2f __attribute__((ext_vector_type(2)));
typedef float v8f __attribute__((ext_vector_type(8)));

#define NPTS        100
#define NPOINT      10
#define BN_MUL_SA   0.9999950000374997f   // 1/sqrt(1+1e-5)
#define BN_MUL_PFN  0.9995003746877732f   // 1/sqrt(1+1e-3)

struct PillarParams {
  const float* features;   // (P,100,4)
  const int*   num_voxels; // (P,)
  const int*   coors;      // (P,4)
  const float* pfn_w; const float* pfn_g; const float* pfn_b;   // (128,9),(128),(128)
  const float* sa0_w[3][4]; const float* sa0_g[3][4]; const float* sa0_b[3][4];
  const float* sa1_w[3];    const float* sa1_g[3];    const float* sa1_b[3];
  float* out;              // (P,128)
};

// ---------------------------------------------------------------------------
// Fused GEMM + BN + ReLU on CDNA5 WMMA (fp32 16x16x4), fully shape-templated.
//   Y[m, n] = relu( (sum_k X[m,k] * W[n,k]) * g[n]*bnmul + b[n] )
// X in LDS, row stride LDX = 4*ceil(K/4) (pad cols zeroed by caller).
// W in global, row-major (COUT x K). Y -> LDS, row stride COUT.
// M register-blocked: MT accumulators live across K; one B fragment feeds MT
// WMMAs. Waves stripe over N tiles; all WMMA control flow is wave-uniform.
// ---------------------------------------------------------------------------
template <int MT, int K, int COUT>
__device__ __forceinline__ void mlp_layer(const float* __restrict__ W,
                                          const float* __restrict__ gamma,
                                          const float* __restrict__ beta,
                                          const float* xb, float* yb,
                                          float bnmul) {
  constexpr int KF  = K >> 2;                  // full k-steps (no masking)
  constexpr int LDX = (K + 3) & ~3;
  constexpr int NT  = COUT >> 4;
  const int wave   = threadIdx.x >> 5;
  const int lane   = threadIdx.x & 31;
  const int mrow   = lane & 15;
  const int khalf  = (lane >> 4) << 1;   // K offset 0 (lanes 0-15) / 2 (16-31)
  const int rshift = (lane >> 4) << 3;   // C/D row offset 0 / 8

  // Warm L2 with the weight matrix (global_prefetch_b8).
  for (int i = (int)threadIdx.x * 32; i < COUT * K; i += 256 * 32)
    __builtin_prefetch(W + i, 0, 0);

  for (int nt = wave; nt < NT; nt += 8) {
    const int col = (nt << 4) + mrow;
    const float gs = gamma[col] * bnmul;
    const float bt = beta[col];
    const float* wrow = W + (size_t)col * K;
    const float* xcol = xb + (size_t)mrow * LDX;

    v8f acc[MT];
#pragma unroll
    for (int m = 0; m < MT; ++m)
      acc[m] = (v8f){0.f, 0.f, 0.f, 0.f, 0.f, 0.f, 0.f, 0.f};

    for (int kt = 0; kt < KF; ++kt) {
      const int kb = (kt << 2) + khalf;
      v2f bf;
      bf.x = wrow[kb];
      bf.y = wrow[kb + 1];
#pragma unroll
      for (int m = 0; m < MT; ++m) {
        const v2f a = *(const v2f*)(xcol + (m << 4) * LDX + kb);  // ds_load_b64
        acc[m] = __builtin_amdgcn_wmma_f32_16x16x4_f32(
            false, a, false, bf, (short)0, acc[m], false, false);
      }
    }
    if constexpr ((K & 3) != 0) {          // single masked tail step, branchless
      const int kb = (KF << 2) + khalf;
      const int i0 = (kb < K) ? kb : 0;
      const int i1 = (kb + 1 < K) ? kb + 1 : 0;
      const float w0 = wrow[i0];
      const float w1 = wrow[i1];
      v2f bf;
      bf.x = (kb     < K) ? w0 : 0.f;
      bf.y = (kb + 1 < K) ? w1 : 0.f;
#pragma unroll
      for (int m = 0; m < MT; ++m) {
        const v2f a = *(const v2f*)(xcol + (m << 4) * LDX + kb);
        acc[m] = __builtin_amdgcn_wmma_f32_16x16x4_f32(
            false, a, false, bf, (short)0, acc[m], false, false);
      }
    }

#pragma unroll
    for (int m = 0; m < MT; ++m) {
      float* ybase = yb + (size_t)((m << 4) + rshift) * COUT + col;
#pragma unroll
      for (int v = 0; v < 8; ++v)
        ybase[(size_t)v * COUT] = fmaxf(acc[m][v] * gs + bt, 0.f);
    }
  }
}

// max over rows [0,R) of an LDS matrix (row stride C), write C values to global
__device__ __forceinline__ void reduce_max_rows(const float* yb, int R, int C,
                                                float* __restrict__ outp) {
  for (int c = threadIdx.x; c < C; c += blockDim.x) {
    float m = -3.4e38f;
    for (int r = 0; r < R; ++r) m = fmaxf(m, yb[(size_t)r * C + c]);
    outp[c] = m;
  }
}

// ---------------------------------------------------------------------------
// One MSG scale: ball query -> group -> 4-layer WMMA MLP -> max over samples.
// ---------------------------------------------------------------------------
template <int SC, int NS, int MT, int C0, int C1, int C2, int C3, int MOFF>
__device__ __forceinline__ void run_scale(const PillarParams& prm, float r2,
                                          float* sX, float* sBuf0, float* sBuf1,
                                          float* sSa1x, int* sBall,
                                          const float* sNewXyz) {
  const int tid = threadIdx.x;
  // ---- ball query: first NS in-range indices (ascending), pad with first --
  if (tid < NPOINT) {
    const float c0 = sNewXyz[tid * 3 + 0];
    const float c1 = sNewXyz[tid * 3 + 1];
    const float c2 = sNewXyz[tid * 3 + 2];
    int cnt = 0;
    for (int j = 0; j < NPTS; ++j) {
      const float dx = sX[j * 12 + 0] - c0;
      const float dy = sX[j * 12 + 1] - c1;
      const float dz = sX[j * 12 + 2] - c2;
      if (dx * dx + dy * dy + dz * dz <= r2) {
        if (cnt < NS) sBall[tid * 10 + cnt] = j;
        ++cnt;
      }
    }
    if (cnt == 0) { sBall[tid * 10] = 0; cnt = 1; }  // unreachable guard
    const int first = sBall[tid * 10];
    for (int k = (cnt < NS ? cnt : NS); k < NS; ++k) sBall[tid * 10 + k] = first;
  }
  __syncthreads();
  // ---- grouped input: rows = centroid*NS + s, 9 ch, K padded to 12 --------
  constexpr int R  = NPOINT * NS;
  constexpr int Rp = MT * 16;
  for (int r = tid; r < Rp; r += 256) {
    float* row = &sBuf0[r * 12];
    if (r < R) {
      const int c  = r / NS;
      const int s2 = r - c * NS;
      const int j  = sBall[c * 10 + s2];
      row[0] = sX[j * 12 + 3];  row[1] = sX[j * 12 + 4];
      row[2] = sX[j * 12 + 5];  row[3] = sX[j * 12 + 6];
      row[4] = sX[j * 12 + 7];  row[5] = sX[j * 12 + 8];
      row[6] = sX[j * 12 + 0] - sNewXyz[c * 3 + 0];
      row[7] = sX[j * 12 + 1] - sNewXyz[c * 3 + 1];
      row[8] = sX[j * 12 + 2] - sNewXyz[c * 3 + 2];
      row[9] = 0.f; row[10] = 0.f; row[11] = 0.f;
    } else {
#pragma unroll
      for (int c = 0; c < 12; ++c) row[c] = 0.f;
    }
  }
  __syncthreads();
  // ---- 4-layer MLP, LDS ping-pong buf0->buf1->buf0->buf1->buf0 ------------
  mlp_layer<MT,  9, C0>(prm.sa0_w[SC][0], prm.sa0_g[SC][0], prm.sa0_b[SC][0],
                        sBuf0, sBuf1, BN_MUL_SA);
  __syncthreads();
  mlp_layer<MT, C0, C1>(prm.sa0_w[SC][1], prm.sa0_g[SC][1], prm.sa0_b[SC][1],
                        sBuf1, sBuf0, BN_MUL_SA);
  __syncthreads();
  mlp_layer<MT, C1, C2>(prm.sa0_w[SC][2], prm.sa0_g[SC][2], prm.sa0_b[SC][2],
                        sBuf0, sBuf1, BN_MUL_SA);
  __syncthreads();
  mlp_layer<MT, C2, C3>(prm.sa0_w[SC][3], prm.sa0_g[SC][3], prm.sa0_b[SC][3],
                        sBuf1, sBuf0, BN_MUL_SA);
  __syncthreads();
  // ---- max over samples per centroid -> msg slice of SA1 input ------------
  for (int e = tid; e < NPOINT * C3; e += 256) {
    const int c  = e / C3;
    const int ch = e - c * C3;
    float m = -3.4e38f;
    for (int s2 = 0; s2 < NS; ++s2)
      m = fmaxf(m, sBuf0[(size_t)(c * NS + s2) * C3 + ch]);
    sSa1x[c * 260 + 3 + MOFF + ch] = m;
  }
  __syncthreads();
}

// ---------------------------------------------------------------------------
// One workgroup (256 threads = 8 wave32) per pillar.
// ---------------------------------------------------------------------------
__global__ __launch_bounds__(256) void pillar_msg_kernel(PillarParams prm) {
  __shared__ float sX[112 * 12];        // 9-ch features, K padded to 12
  __shared__ float sBuf0[112 * 128];    // activation ping
  __shared__ float sBuf1[112 * 128];    // activation pong
  __shared__ float sSa1x[16 * 260];     // SA1 input: [xyz(3) | msg(256) | pad]
  __shared__ float sDist[112];
  __shared__ float sNewXyz[NPOINT * 3];
  __shared__ float sMean[3];
  __shared__ int   sSel[NPOINT];
  __shared__ int   sBall[NPOINT * 10];
  __shared__ int   sFarthest;

  const int p   = blockIdx.x;
  const int tid = threadIdx.x;
  const int nv  = prm.num_voxels[p];
  float* out = prm.out + (size_t)p * 128;

  // ---- Phase 1: stage raw features, compute mean, build 9-ch masked feats --
  const float* fp = prm.features + (size_t)p * NPTS * 4;
  for (int n = tid; n < NPTS; n += 256) {
    const float4 r = ((const float4*)fp)[n];
    sX[n * 12 + 0] = r.x; sX[n * 12 + 1] = r.y;
    sX[n * 12 + 2] = r.z; sX[n * 12 + 3] = r.w;
  }
  __syncthreads();
  if (tid < 3) {  // points_mean: sum over ALL 100 rows / nv (matches reference)
    float s = 0.f;
    for (int n = 0; n < NPTS; ++n) s += sX[n * 12 + tid];
    sMean[tid] = s / (float)nv;
  }
  __syncthreads();
  const float cx = (float)prm.coors[p * 4 + 3] * 0.2f + 0.1f;    // VX/2 + 0
  const float cy = (float)prm.coors[p * 4 + 2] * 0.2f - 39.9f;   // VY/2 - 40
  for (int n = tid; n < 112; n += 256) {
    float* row = &sX[n * 12];
    if (n < NPTS) {
      const float r0 = row[0], r1 = row[1], r2 = row[2], r3 = row[3];
      const float m = (n < nv) ? 1.f : 0.f;
      row[0] = r0 * m; row[1] = r1 * m; row[2] = r2 * m; row[3] = r3 * m;
      row[4] = (r0 - sMean[0]) * m;
      row[5] = (r1 - sMean[1]) * m;
      row[6] = (r2 - sMean[2]) * m;
      row[7] = (r0 - cx) * m;
      row[8] = (r1 - cy) * m;
      row[9] = 0.f; row[10] = 0.f; row[11] = 0.f;
    } else {
#pragma unroll
      for (int c = 0; c < 12; ++c) row[c] = 0.f;   // M-pad rows
    }
  }
  __syncthreads();

  // ========================== PFN path (nv <= 50) ==========================
  if (nv <= 50) {
    mlp_layer<7, 9, 128>(prm.pfn_w, prm.pfn_g, prm.pfn_b, sX, sBuf0, BN_MUL_PFN);
    __syncthreads();
    reduce_max_rows(sBuf0, NPTS, 128, out);
    return;
  }

  // ======================= PointNet++ MSG path (nv > 50) ===================
  // ---- Farthest Point Sampling (10 centroids, first-max tie-break) --------
  if (tid < NPTS) sDist[tid] = 1e10f;
  if (tid == 0) sFarthest = 0;
  __syncthreads();
  for (int it = 0; it < NPOINT; ++it) {
    const int fa = sFarthest;
    if (tid == 0) sSel[it] = fa;
    const float c0 = sX[fa * 12 + 0], c1 = sX[fa * 12 + 1], c2 = sX[fa * 12 + 2];
    if (tid < NPTS) {
      const float dx = sX[tid * 12 + 0] - c0;
      const float dy = sX[tid * 12 + 1] - c1;
      const float dz = sX[tid * 12 + 2] - c2;
      sDist[tid] = fminf(sDist[tid], dx * dx + dy * dy + dz * dz);
    }
    __syncthreads();
    if (tid == 0) {
      float best = -1.f; int bi = 0;
      for (int j = 0; j < NPTS; ++j) {
        const float v = sDist[j];
        if (v > best) { best = v; bi = j; }
      }
      sFarthest = bi;
    }
    __syncthreads();
  }
  if (tid < NPOINT) {
    const int s = sSel[tid];
    sNewXyz[tid * 3 + 0] = sX[s * 12 + 0];
    sNewXyz[tid * 3 + 1] = sX[s * 12 + 1];
    sNewXyz[tid * 3 + 2] = sX[s * 12 + 2];
  }
  for (int e = tid; e < 16 * 260; e += 256) sSa1x[e] = 0.f;
  __syncthreads();
  if (tid < NPOINT) {
    sSa1x[tid * 260 + 0] = sNewXyz[tid * 3 + 0];
    sSa1x[tid * 260 + 1] = sNewXyz[tid * 3 + 1];
    sSa1x[tid * 260 + 2] = sNewXyz[tid * 3 + 2];
  }
  __syncthreads();

  // ---- three MSG scales (shapes compile-time) -----------------------------
  run_scale<0,  6, 4, 16, 32, 48,  64,   0>(prm, (float)(0.32 * 0.32),
      sX, sBuf0, sBuf1, sSa1x, sBall, sNewXyz);
  run_scale<1,  8, 5, 32, 32, 48,  64,  64>(prm, (float)(0.36 * 0.36),
      sX, sBuf0, sBuf1, sSa1x, sBall, sNewXyz);
  run_scale<2, 10, 7, 32, 64, 96, 128, 128>(prm, (float)(0.40 * 0.40),
      sX, sBuf0, sBuf1, sSa1x, sBall, sNewXyz);

  // ---- SA1 group-all MLP: (10,259)->512->256->128, then max over 10 rows --
  mlp_layer<1, 259, 512>(prm.sa1_w[0], prm.sa1_g[0], prm.sa1_b[0],
                         sSa1x, sBuf0, BN_MUL_SA);
  __syncthreads();
  mlp_layer<1, 512, 256>(prm.sa1_w[1], prm.sa1_g[1], prm.sa1_b[1],
                         sBuf0, sBuf1, BN_MUL_SA);
  __syncthreads();
  mlp_layer<1, 256, 128>(prm.sa1_w[2], prm.sa1_g[2], prm.sa1_b[2],
                         sBuf1, sBuf0, BN_MUL_SA);
  __syncthreads();
  reduce_max_rows(sBuf0, NPOINT, 128, out);
}

// ---------------------------------------------------------------------------
extern "C" void kernel_launch(void* const* d_in, const int* in_sizes, int n_in,
                              void* d_out, int out_size, void* d_ws, size_t ws_size,
                              hipStream_t stream) {
  (void)d_ws; (void)ws_size; (void)out_size;
  static const int SA0_DIMS[3][5] = {{9, 16, 32, 48, 64},
                                     {9, 32, 32, 48, 64},
                                     {9, 32, 64, 96, 128}};
  static const int SA1_DIMS[4] = {259, 512, 256, 128};

  PillarParams prm;
  prm.features   = (const float*)d_in[0];
  prm.num_voxels = (const int*)d_in[1];
  prm.coors      = (const int*)d_in[2];
  prm.pfn_w      = (const float*)d_in[3];
  prm.pfn_g      = (const float*)d_in[4];
  prm.pfn_b      = (const float*)d_in[5];

  if (n_in >= 51) {
    // pytree leaves: per layer (W, g, b) in order
    int t = 6;
    for (int sc = 0; sc < 3; ++sc)
      for (int L = 0; L < 4; ++L) {
        prm.sa0_w[sc][L] = (const float*)d_in[t++];
        prm.sa0_g[sc][L] = (const float*)d_in[t++];
        prm.sa0_b[sc][L] = (const float*)d_in[t++];
      }
    for (int L = 0; L < 3; ++L) {
      prm.sa1_w[L] = (const float*)d_in[t++];
      prm.sa1_g[L] = (const float*)d_in[t++];
      prm.sa1_b[L] = (const float*)d_in[t++];
    }
  } else {
    // fallback: sa0_params / sa1_params each packed flat in leaf order
    const float* base = (const float*)d_in[6];
    size_t off = 0;
    for (int sc = 0; sc < 3; ++sc)
      for (int L = 0; L < 4; ++L) {
        const int cin = SA0_DIMS[sc][L], cout = SA0_DIMS[sc][L + 1];
        prm.sa0_w[sc][L] = base + off; off += (size_t)cout * cin;
        prm.sa0_g[sc][L] = base + off; off += cout;
        prm.sa0_b[sc][L] = base + off; off += cout;
      }
    base = (const float*)d_in[7];
    off = 0;
    for (int L = 0; L < 3; ++L) {
      const int cin = SA1_DIMS[L], cout = SA1_DIMS[L + 1];
      prm.sa1_w[L] = base + off; off += (size_t)cout * cin;
      prm.sa1_g[L] = base + off; off += cout;
      prm.sa1_b[L] = base + off; off += cout;
    }
  }
  prm.out = (float*)d_out;

  const int P = in_sizes[1];   // number of pillars (num_voxels count)
  pillar_msg_kernel<<<P, 256, 0, stream>>>(prm);
}